// S4_48189533061842
// MI455X (gfx1250) — compile-verified
//
#include <hip/hip_runtime.h>

// S4 SSM: ys = scan(Ab, Bb@u) @ C.T with bilinear discretization.
// Strategy (MI455X, wave32, fp32 WMMA 16x16x4 for exact fp32 GEMMs):
//  1. M = A*step/2; BL = inv(I-M) via Neumann/Horner (||M||~1e-4 -> 4 GEMMs).
//  2. Ab = BL@(I+M); Bb = step*(BL@B); P = Ab^64 via 6 squarings.
//  3. Bu = u @ Bb.T  (16384x512x512 WMMA GEMM).
//  4. Chunked scan: 256 chunks of T=64. Pass1: per-WG batched local scans
//     (state in LDS, 64 sequential WMMA GEMM steps, no global sync).
//  5. Boundary scan over 256 chunks with Ab^64 (single WG).
//  6. Pass2: rerun chunk scan seeded with boundary states -> exact xs.
//  7. ys = xs @ C.T (WMMA GEMM) -> d_out.

typedef float v2f __attribute__((ext_vector_type(2)));
typedef float v8f __attribute__((ext_vector_type(8)));

#define N_S     512
#define D_F     512
#define L_SEQ   16384
#define T_CH    64
#define N_CHUNK (L_SEQ / T_CH)   // 256
#define CB      16               // chunk-columns per workgroup (WMMA N dim)
#define CBP     17               // padded LDS stride (bank-conflict avoidance)
#define SCAN_WAVES 8
#define SCAN_MT (N_S / 16 / SCAN_WAVES) // 4 m-tiles per wave

__device__ __forceinline__ v8f wmma_f32(v2f a, v2f b, v8f c) {
  // D = A(16x4,f32) x B(4x16,f32) + C(16x16,f32), exact fp32
  return __builtin_amdgcn_wmma_f32_16x16x4_f32(false, a, false, b,
                                               (short)0, c, false, false);
}

// ---------------------------------------------------------------------------
// M = A*h ; IpM = I + A*h
__global__ void build_M_kernel(const float* __restrict__ A,
                               float* __restrict__ Mh,
                               float* __restrict__ IpM, float h) {
  int i = blockIdx.x * blockDim.x + threadIdx.x;   // exactly N_S*N_S threads
  float m = A[i] * h;
  Mh[i] = m;
  int r = i >> 9, c = i & (N_S - 1);
  IpM[i] = m + (r == c ? 1.0f : 0.0f);
}

// ---------------------------------------------------------------------------
// C[M x Nc] = alpha * A[M x K] @ B  (+ I if ADDI).
// TRANSB: B stored row-major (Nc x K), used as B^T (logical K x Nc).
// One wave computes a 16x32 tile (two 16x16 WMMA accumulators sharing A frag).
template <bool TRANSB, bool ADDI>
__global__ void gemm_kernel(const float* __restrict__ A,
                            const float* __restrict__ B,
                            float* __restrict__ C,
                            int M, int Nc, int K, float alpha) {
  const int lane   = threadIdx.x & 31;
  const int wave   = threadIdx.x >> 5;
  const int lane15 = lane & 15;
  const int hi     = lane >> 4;     // lanes 16-31 hold K=2,3 (A) / rows +8 (C)
  const int koff   = hi * 2;

  const int ntn  = Nc / 32;
  const int tile = blockIdx.x * 8 + wave;
  const int mb   = (tile / ntn) * 16;
  const int nb   = (tile % ntn) * 32;

  v8f acc0 = {}, acc1 = {};
  const float* Arow = A + (size_t)(mb + lane15) * K;
  for (int kb = 0; kb < K; kb += 4) {
    v2f a = *(const v2f*)(Arow + kb + koff);
    v2f b0, b1;
    if (TRANSB) {
      b0 = *(const v2f*)(B + (size_t)(nb + lane15) * K + kb + koff);
      b1 = *(const v2f*)(B + (size_t)(nb + 16 + lane15) * K + kb + koff);
    } else {
      b0.x = B[(size_t)(kb + koff)     * Nc + nb + lane15];
      b0.y = B[(size_t)(kb + koff + 1) * Nc + nb + lane15];
      b1.x = B[(size_t)(kb + koff)     * Nc + nb + 16 + lane15];
      b1.y = B[(size_t)(kb + koff + 1) * Nc + nb + 16 + lane15];
    }
    acc0 = wmma_f32(a, b0, acc0);
    acc1 = wmma_f32(a, b1, acc1);
  }
#pragma unroll
  for (int v = 0; v < 8; ++v) {
    int row = mb + v + hi * 8;
    int c0  = nb + lane15;
    int c1  = nb + 16 + lane15;
    float r0 = acc0[v] * alpha;
    float r1 = acc1[v] * alpha;
    if (ADDI) {
      if (row == c0) r0 += 1.0f;
      if (row == c1) r1 += 1.0f;
    }
    C[(size_t)row * Nc + c0] = r0;
    C[(size_t)row * Nc + c1] = r1;
  }
}

// ---------------------------------------------------------------------------
// Batched chunk scan. Each WG owns CB=16 chunk columns; state X (512 x 16)
// lives in LDS. Per step: X = Ab @ X + Bu_slice via fp32 WMMA (32 m-tiles
// split over 8 waves). PASS==1: X starts at 0, emit chunk-final states F.
// PASS==2: X starts at boundary state Xb, emit full xs.
template <int PASS>
__global__ void scan_kernel(const float* __restrict__ Ab,
                            const float* __restrict__ Bu,
                            const float* __restrict__ Xb,
                            float* __restrict__ F,
                            float* __restrict__ xs) {
  __shared__ float Xs[N_S * CBP];
  const int tid    = threadIdx.x;
  const int lane15 = tid & 15;
  const int hi     = (tid >> 4) & 1;
  const int wave   = tid >> 5;
  const int koff   = hi * 2;
  const int chunk0 = blockIdx.x * CB;
  const int mychunk = chunk0 + lane15;

  for (int i = tid; i < N_S * CB; i += 256) {
    int k = i / CB, c = i % CB;
    Xs[k * CBP + c] =
        (PASS == 1) ? 0.0f : Xb[(size_t)k * N_CHUNK + chunk0 + c];
  }
  __syncthreads();

  v8f acc[SCAN_MT];
  for (int t = 0; t < T_CH; ++t) {
    const size_t l = (size_t)mychunk * T_CH + t;
    // accumulators seeded with this step's Bu fragment
#pragma unroll
    for (int j = 0; j < SCAN_MT; ++j) {
      int mb = (wave * SCAN_MT + j) * 16;
#pragma unroll
      for (int v = 0; v < 8; ++v)
        acc[j][v] = Bu[l * N_S + mb + v + hi * 8];
    }
    for (int kb = 0; kb < N_S; kb += 4) {
      v2f b;  // shared across the wave's 4 m-tiles
      b.x = Xs[(kb + koff)     * CBP + lane15];
      b.y = Xs[(kb + koff + 1) * CBP + lane15];
#pragma unroll
      for (int j = 0; j < SCAN_MT; ++j) {
        int row = (wave * SCAN_MT + j) * 16 + lane15;
        v2f a = *(const v2f*)(Ab + (size_t)row * N_S + kb + koff);
        acc[j] = wmma_f32(a, b, acc[j]);
      }
    }
    __syncthreads();  // all reads of X done before overwrite
#pragma unroll
    for (int j = 0; j < SCAN_MT; ++j) {
      int mb = (wave * SCAN_MT + j) * 16;
#pragma unroll
      for (int v = 0; v < 8; ++v) {
        int row = mb + v + hi * 8;
        Xs[row * CBP + lane15] = acc[j][v];
        if (PASS == 2) xs[l * N_S + row] = acc[j][v];
      }
    }
    __syncthreads();
  }
  if (PASS == 1) {
#pragma unroll
    for (int j = 0; j < SCAN_MT; ++j) {
      int mb = (wave * SCAN_MT + j) * 16;
#pragma unroll
      for (int v = 0; v < 8; ++v)
        F[(size_t)(mb + v + hi * 8) * N_CHUNK + mychunk] = acc[j][v];
    }
  }
}

// ---------------------------------------------------------------------------
// Boundary scan: xb_{i+1} = Ab^T xb_i + F_i, 256 sequential 512x512 matvecs.
// Xb[:, i] = state at the START of chunk i.  Single WG of 512 threads.
__global__ void boundary_kernel(const float* __restrict__ P,   // Ab^64
                                const float* __restrict__ F,
                                float* __restrict__ Xb) {
  __shared__ float xc[N_S];
  const int r = threadIdx.x;
  xc[r] = 0.0f;
  __syncthreads();
  const float* Prow = P + (size_t)r * N_S;
  for (int i = 0; i < N_CHUNK; ++i) {
    Xb[(size_t)r * N_CHUNK + i] = xc[r];
    float s = F[(size_t)r * N_CHUNK + i];
    for (int k = 0; k < N_S; k += 4) {
      const float4 p = *(const float4*)(Prow + k);
      s += p.x * xc[k] + p.y * xc[k + 1] + p.z * xc[k + 2] + p.w * xc[k + 3];
    }
    __syncthreads();   // all reads of xc complete
    xc[r] = s;
    __syncthreads();
  }
}

// ---------------------------------------------------------------------------
extern "C" void kernel_launch(void* const* d_in, const int* in_sizes, int n_in,
                              void* d_out, int out_size, void* d_ws,
                              size_t ws_size, hipStream_t stream) {
  (void)in_sizes; (void)n_in; (void)out_size; (void)ws_size;
  const float* A = (const float*)d_in[0];   // [N, N]
  const float* B = (const float*)d_in[1];   // [N, D]
  const float* C = (const float*)d_in[2];   // [D, N]
  const float* u = (const float*)d_in[3];   // [L, D]
  float* ys = (float*)d_out;                // [L, D]

  float* ws = (float*)d_ws;
  const size_t NN = (size_t)N_S * N_S;
  float* Mh  = ws;
  float* IpM = Mh  + NN;
  float* t0  = IpM + NN;
  float* t1  = t0  + NN;
  float* BL  = t1  + NN;
  float* Ab  = BL  + NN;
  float* Pa  = Ab  + NN;
  float* Pb  = Pa  + NN;
  float* Bb  = Pb  + NN;
  float* Bu  = Bb  + NN;                         // [L, N]
  float* F   = Bu  + (size_t)L_SEQ * N_S;        // [N, N_CHUNK]
  float* Xb  = F   + (size_t)N_S * N_CHUNK;      // [N, N_CHUNK]
  float* xs  = Xb  + (size_t)N_S * N_CHUNK;      // [L, N]

  const float step = 1.0f / (float)L_SEQ;
  const float h    = step * 0.5f;

  build_M_kernel<<<NN / 256, 256, 0, stream>>>(A, Mh, IpM, h);

#define GEMM(TB, AI, Ap, Bp, Cp, M_, Nc_, K_, al_)                         \
  gemm_kernel<TB, AI><<<((M_ / 16) * (Nc_ / 32)) / 8, 256, 0, stream>>>(   \
      Ap, Bp, Cp, M_, Nc_, K_, al_)

  // BL = inv(I - M) via Horner: S <- M@S + I  (starting S = I + M)
  GEMM(false, true,  Mh, IpM, t0, N_S, N_S, N_S, 1.0f);
  GEMM(false, true,  Mh, t0,  t1, N_S, N_S, N_S, 1.0f);
  GEMM(false, true,  Mh, t1,  t0, N_S, N_S, N_S, 1.0f);
  GEMM(false, true,  Mh, t0,  BL, N_S, N_S, N_S, 1.0f);
  // Ab = BL @ (I + M); Bb = step * (BL @ B)
  GEMM(false, false, BL, IpM, Ab, N_S, N_S, N_S, 1.0f);
  GEMM(false, false, BL, B,   Bb, N_S, D_F, N_S, step);
  // Pb = Ab^64 by repeated squaring
  GEMM(false, false, Ab, Ab,  Pa, N_S, N_S, N_S, 1.0f);   // Ab^2
  GEMM(false, false, Pa, Pa,  Pb, N_S, N_S, N_S, 1.0f);   // Ab^4
  GEMM(false, false, Pb, Pb,  Pa, N_S, N_S, N_S, 1.0f);   // Ab^8
  GEMM(false, false, Pa, Pa,  Pb, N_S, N_S, N_S, 1.0f);   // Ab^16
  GEMM(false, false, Pb, Pb,  Pa, N_S, N_S, N_S, 1.0f);   // Ab^32
  GEMM(false, false, Pa, Pa,  Pb, N_S, N_S, N_S, 1.0f);   // Ab^64
  // Bu = u @ Bb.T
  GEMM(true,  false, u,  Bb,  Bu, L_SEQ, N_S, D_F, 1.0f);

  scan_kernel<1><<<N_CHUNK / CB, 256, 0, stream>>>(Ab, Bu, nullptr, F, nullptr);
  boundary_kernel<<<1, N_S, 0, stream>>>(Pb, F, Xb);
  scan_kernel<2><<<N_CHUNK / CB, 256, 0, stream>>>(Ab, Bu, Xb, nullptr, xs);

  // ys = xs @ C.T
  GEMM(true, false, xs, C, ys, L_SEQ, D_F, N_S, 1.0f);
#undef GEMM
}